// ExpertsText_16896401343011
// MI455X (gfx1250) — compile-verified
//
#include <hip/hip_runtime.h>
#include <math.h>

// Problem constants (match reference)
#define BB   4
#define SS   2048
#define EMBD 1024
#define HIDD 2048
#define NE   8
#define NTOK (BB * SS)      // 8192 tokens
#define NROW (NTOK * 2)     // 16384 (token,slot) rows
#define LDA  1028           // LDS A stride: 1028 % 64 == 4 -> conflict-free frags
#define MAX_RT 1032         // worst-case sum of ceil(count[e]/16)

typedef float v2f __attribute__((ext_vector_type(2)));
typedef float v4f __attribute__((ext_vector_type(4)));
typedef float v8f __attribute__((ext_vector_type(8)));

// Workspace layout (as int32 words):
//  [0..7]    counts[e]
//  [8..15]   cursor[e]
//  [16..24]  offsets[e] (exclusive prefix of counts, +total)
//  [32..40]  tileStart[e] (prefix of ceil(counts/16), +total)
//  [64..64+NROW)          topIdx  (expert id per (token,slot))
//  [64+NROW..64+2*NROW)   rowIndex (gathered row codes, grouped by expert)

__global__ void __launch_bounds__(32) moe_init(int* wsI) {
    if (threadIdx.x < 16) wsI[threadIdx.x] = 0;   // counts + cursors
}

// --- Gating: one wave per token: logits -> softmax -> top2 ---
__global__ void __launch_bounds__(256) moe_gating(const float* __restrict__ x,
                                                  const float* __restrict__ Wg,
                                                  const float* __restrict__ bg,
                                                  float* __restrict__ out,
                                                  int* __restrict__ wsI) {
    const int wave = threadIdx.x >> 5;
    const int lane = threadIdx.x & 31;
    const int t = blockIdx.x * 8 + wave;

    const float* xr = x + (size_t)t * EMBD;
    float acc[NE] = {0.f, 0.f, 0.f, 0.f, 0.f, 0.f, 0.f, 0.f};

#pragma unroll 4
    for (int j = 0; j < EMBD / 32; ++j) {
        const int k = j * 32 + lane;             // coalesced across lanes
        const float xv = xr[k];
        const v4f* wrow = (const v4f*)(Wg + (size_t)k * NE);
        const v4f w0 = wrow[0];
        const v4f w1 = wrow[1];
        acc[0] += xv * w0.x; acc[1] += xv * w0.y;
        acc[2] += xv * w0.z; acc[3] += xv * w0.w;
        acc[4] += xv * w1.x; acc[5] += xv * w1.y;
        acc[6] += xv * w1.z; acc[7] += xv * w1.w;
    }
    // wave32 butterfly reduction
#pragma unroll
    for (int e = 0; e < NE; ++e)
#pragma unroll
        for (int off = 16; off >= 1; off >>= 1)
            acc[e] += __shfl_xor(acc[e], off, 32);

    if (lane == 0) {
        float l[NE], w[NE];
        float mx = -1e30f;
#pragma unroll
        for (int e = 0; e < NE; ++e) { l[e] = acc[e] + bg[e]; mx = fmaxf(mx, l[e]); }
        float s = 0.f;
#pragma unroll
        for (int e = 0; e < NE; ++e) { w[e] = expf(l[e] - mx); s += w[e]; }
        const float inv = 1.f / s;
#pragma unroll
        for (int e = 0; e < NE; ++e) w[e] *= inv;
        // top-2, ties -> lowest index first (matches jax.lax.top_k)
        int e0 = 0;
#pragma unroll
        for (int e = 1; e < NE; ++e) if (w[e] > w[e0]) e0 = e;
        int e1 = -1;
#pragma unroll
        for (int e = 0; e < NE; ++e)
            if (e != e0 && (e1 < 0 || w[e] > w[e1])) e1 = e;

        out[t * 2 + 0] = w[e0];
        out[t * 2 + 1] = w[e1];
        wsI[64 + t * 2 + 0] = e0;
        wsI[64 + t * 2 + 1] = e1;
        atomicAdd(&wsI[e0], 1);
        atomicAdd(&wsI[e1], 1);
    }
}

// --- Tiny prefix scan over 8 experts ---
__global__ void __launch_bounds__(32) moe_prefix(int* wsI) {
    if (threadIdx.x == 0) {
        int off = 0, ts = 0;
        for (int e = 0; e < NE; ++e) {
            wsI[16 + e] = off;
            wsI[32 + e] = ts;
            off += wsI[e];
            ts  += (wsI[e] + 15) >> 4;
        }
        wsI[16 + NE] = off;
        wsI[32 + NE] = ts;
    }
}

// --- Scatter (token,slot) row codes into expert buckets ---
__global__ void __launch_bounds__(256) moe_scatter(int* wsI) {
    const int i = blockIdx.x * 256 + threadIdx.x;   // row code = token*2 + slot
    const int e = wsI[64 + i];
    const int pos = atomicAdd(&wsI[8 + e], 1);
    wsI[64 + NROW + wsI[16 + e] + pos] = i;
}

// --- Grouped GEMM: per block, 16 gathered rows x 128 cols of one expert ---
__global__ void __launch_bounds__(256) moe_gemm(const float* __restrict__ x,
                                                const float* __restrict__ We,
                                                const float* __restrict__ be,
                                                float* __restrict__ out,
                                                const int* __restrict__ wsI) {
    __shared__ float As[16 * LDA];
    __shared__ int sRow[16];
    __shared__ int sVal[16];

    const int rt = blockIdx.x;
    const int totalTiles = wsI[32 + NE];
    if (rt >= totalTiles) return;

    // locate expert bucket: tileStart[e] <= rt < tileStart[e+1]
    int e = 0;
    while (e < NE - 1 && rt >= wsI[32 + e + 1]) ++e;
    const int lt   = rt - wsI[32 + e];
    const int cnt  = wsI[e];
    const int base = wsI[16 + e];
    const int* rowIdx = wsI + 64 + NROW;

    const int tid = threadIdx.x;
    if (tid < 16) {
        const int r = lt * 16 + tid;
        const int valid = (r < cnt) ? 1 : 0;
        sRow[tid] = valid ? rowIdx[base + r] : rowIdx[base + lt * 16];
        sVal[tid] = valid;
    }
    __syncthreads();

    // stage A tile (16 x 1024 f32) into LDS, float4-vectorized, coalesced
#pragma unroll 4
    for (int i = tid; i < 16 * (EMBD / 4); i += 256) {
        const int row = i >> 8;            // EMBD/4 = 256 float4 per row
        const int c4  = i & 255;
        const int tok = sRow[row] >> 1;
        const v4f v = *((const v4f*)(x + (size_t)tok * EMBD) + c4);
        *(v4f*)(&As[row * LDA + c4 * 4]) = v;
    }
    __syncthreads();

    const int wave = tid >> 5;
    const int lane = tid & 31;
    const int half = lane >> 4;            // 0: K+0/1, 1: K+2/3 (A/B frag layout)
    const int ln   = lane & 15;
    const int k0   = 2 * half;
    const int n    = blockIdx.y * 128 + wave * 16 + ln;

    const float* Bp = We + (size_t)e * EMBD * HIDD + n;
    const float* Ap = As + ln * LDA + k0;

    v8f c0 = {};
    v8f c1 = {};
    // Two independent WMMA accumulation chains over K to hide WMMA latency.
#pragma unroll 4
    for (int kk = 0; kk < EMBD; kk += 8) {
        v2f a0 = *(const v2f*)(Ap + kk);
        v2f b0;
        b0.x = Bp[(size_t)(kk + k0 + 0) * HIDD];
        b0.y = Bp[(size_t)(kk + k0 + 1) * HIDD];
        c0 = __builtin_amdgcn_wmma_f32_16x16x4_f32(false, a0, false, b0,
                                                   (short)0, c0, false, false);
        v2f a1 = *(const v2f*)(Ap + kk + 4);
        v2f b1;
        b1.x = Bp[(size_t)(kk + 4 + k0 + 0) * HIDD];
        b1.y = Bp[(size_t)(kk + 4 + k0 + 1) * HIDD];
        c1 = __builtin_amdgcn_wmma_f32_16x16x4_f32(false, a1, false, b1,
                                                   (short)0, c1, false, false);
    }

    const float bias = be[(size_t)e * HIDD + n];
    const size_t outBase = (size_t)NROW;   // expert outputs follow top2 weights
#pragma unroll
    for (int r = 0; r < 8; ++r) {
        const int m = r + 8 * half;        // C/D layout: lanes 16-31 hold M=8..15
        if (sVal[m])
            out[outBase + (size_t)sRow[m] * HIDD + n] = c0[r] + c1[r] + bias;
    }
}

extern "C" void kernel_launch(void* const* d_in, const int* in_sizes, int n_in,
                              void* d_out, int out_size, void* d_ws, size_t ws_size,
                              hipStream_t stream) {
    const float* x  = (const float*)d_in[0];
    const float* Wg = (const float*)d_in[1];
    const float* bg = (const float*)d_in[2];
    const float* We = (const float*)d_in[3];
    const float* be = (const float*)d_in[4];
    float* out = (float*)d_out;
    int* wsI = (int*)d_ws;

    moe_init<<<1, 32, 0, stream>>>(wsI);
    moe_gating<<<NTOK / 8, 256, 0, stream>>>(x, Wg, bg, out, wsI);
    moe_prefix<<<1, 32, 0, stream>>>(wsI);
    moe_scatter<<<NROW / 256, 256, 0, stream>>>(wsI);
    moe_gemm<<<dim3(MAX_RT, HIDD / 128), 256, 0, stream>>>(x, We, be, out, wsI);
}